// LearnedMultiScaleGuidedFilter_78915729097397
// MI455X (gfx1250) — compile-verified
//
#include <hip/hip_runtime.h>
#include <hip/hip_bf16.h>
#include <math.h>

typedef float v2f __attribute__((ext_vector_type(2)));
typedef float v8f __attribute__((ext_vector_type(8)));

#define H_LR 256
#define W_LR 256
#define H_HR 1024
#define W_HR 1024
#define MAX_NS 8
#define LOG_SMALL_EPS_F (-11.512925464970229f)
#define LDS_PITCH 260

struct ScaleParams { int kh[MAX_NS]; int kw[MAX_NS]; int ns; };

// ---------------------------------------------------------------------------
// K1: cumulative sum along W via WMMA:  S = X * U_tri  (U[k,n] = k<=n)
// One wave (32 lanes) per (quantity q, image, 16-row strip).
// Blocked scan: per 16-col tile, 4x V_WMMA_F32_16X16X4_F32 on the triangular
// diagonal block; carry (last column) broadcast via __shfl as next tile's C.
// q: 0=target 1=guide 2=target*guide 3=guide*guide
// ---------------------------------------------------------------------------
__global__ __launch_bounds__(32) void k_cumsum_w(const float* __restrict__ target,
                                                 const float* __restrict__ guide,
                                                 float* __restrict__ II, int bc) {
  int idx = blockIdx.x;
  int strip = idx & 15; idx >>= 4;
  int img = idx % bc;
  int q = idx / bc;
  int lane = threadIdx.x;

  __shared__ float sx[16 * LDS_PITCH];

  const float* t = target + (size_t)img * (H_LR * W_LR) + (size_t)strip * 16 * W_LR;
  const float* g = guide  + (size_t)img * (H_LR * W_LR) + (size_t)strip * 16 * W_LR;
  for (int tix = lane; tix < 16 * W_LR; tix += 32) {
    int r = tix >> 8;        // W_LR == 256
    int c = tix & 255;
    float tv = t[r * W_LR + c];
    float gv = g[r * W_LR + c];
    float v = (q == 0) ? tv : (q == 1) ? gv : (q == 2) ? (tv * gv) : (gv * gv);
    sx[r * LDS_PITCH + c] = v;
  }
  __syncthreads();

  float* out = II + ((size_t)q * bc + img) * (H_LR * W_LR) + (size_t)strip * 16 * W_LR;

  const int mn       = lane & 15;               // A's M index == B's N index
  const int koff     = (lane < 16) ? 0 : 2;     // K offset per half-wave
  const int row_off  = (lane < 16) ? 0 : 8;     // C/D row offset
  const int src_lane = (lane < 16) ? 15 : 31;   // lane holding column N=15

  v8f carry;
  #pragma unroll
  for (int v = 0; v < 8; ++v) carry[v] = 0.0f;

  for (int jt = 0; jt < 16; ++jt) {
    v8f acc = carry;                            // C = carry broadcast over N
    const int cb = jt * 16;
    #pragma unroll
    for (int kk = 0; kk < 4; ++kk) {
      const int k0 = kk * 4;
      v2f a, b;
      // A = data tile: elem(M=mn, K=k0+koff(+1))
      a.x = sx[mn * LDS_PITCH + cb + k0 + koff];
      a.y = sx[mn * LDS_PITCH + cb + k0 + koff + 1];
      // B = upper-tri ones: U[k,n] = (k <= n)
      b.x = ((k0 + koff)     <= mn) ? 1.0f : 0.0f;
      b.y = ((k0 + koff + 1) <= mn) ? 1.0f : 0.0f;
      acc = __builtin_amdgcn_wmma_f32_16x16x4_f32(false, a, false, b,
                                                  (short)0, acc, false, false);
    }
    // store D tile: elem v -> (row v+row_off, col cb+mn)
    #pragma unroll
    for (int v = 0; v < 8; ++v)
      out[(v + row_off) * W_LR + cb + mn] = acc[v];
    // carry[row] = D[row, 15] (held by lanes 15 / 31)
    #pragma unroll
    for (int v = 0; v < 8; ++v)
      carry[v] = __shfl(acc[v], src_lane, 32);
  }
}

// ---------------------------------------------------------------------------
// K2: cumulative sum along H (in place) via WMMA:  S = L_tri * Y
// One wave per (q, image, 16-col tile). Carry = last row of previous strip.
// ---------------------------------------------------------------------------
__global__ __launch_bounds__(32) void k_cumsum_h(float* __restrict__ II, int bc) {
  int idx = blockIdx.x;
  int ct = idx & 15; idx >>= 4;
  int img = idx % bc;
  int q = idx / bc;
  int lane = threadIdx.x;

  float* base = II + ((size_t)q * bc + img) * (H_LR * W_LR);
  const int mn      = lane & 15;
  const int koff    = (lane < 16) ? 0 : 2;
  const int row_off = (lane < 16) ? 0 : 8;
  const int cb = ct * 16;

  float cv = 0.0f;   // carry for this lane's column (same for every row)
  for (int it = 0; it < 16; ++it) {
    v8f acc;
    #pragma unroll
    for (int v = 0; v < 8; ++v) acc[v] = cv;
    const int rb = it * 16;
    #pragma unroll
    for (int kk = 0; kk < 4; ++kk) {
      const int k0 = kk * 4;
      v2f a, b;
      // A = lower-tri ones: L[m,k] = (k <= m)
      a.x = ((k0 + koff)     <= mn) ? 1.0f : 0.0f;
      a.y = ((k0 + koff + 1) <= mn) ? 1.0f : 0.0f;
      // B = data tile: elem(K=k0+koff(+1), N=mn)
      b.x = base[(rb + k0 + koff)     * W_LR + cb + mn];
      b.y = base[(rb + k0 + koff + 1) * W_LR + cb + mn];
      acc = __builtin_amdgcn_wmma_f32_16x16x4_f32(false, a, false, b,
                                                  (short)0, acc, false, false);
    }
    #pragma unroll
    for (int v = 0; v < 8; ++v)
      base[(rb + v + row_off) * W_LR + cb + mn] = acc[v];
    // carry = D row 15 (VGPR 7, lanes 16..31, N = lane-16)
    cv = __shfl(acc[7], 16 + mn, 32);
  }
}

// ---------------------------------------------------------------------------
// K3: per-pixel multi-scale stats from integral images + softmax combine
// ---------------------------------------------------------------------------
__global__ void k_stats(const float* __restrict__ II,
                        const float* __restrict__ log_eps,
                        const float* __restrict__ wts,
                        float* __restrict__ Ac, float* __restrict__ Bc,
                        int bc, ScaleParams sp) {
  int p = blockIdx.x * blockDim.x + threadIdx.x;
  int total = bc * H_LR * W_LR;
  if (p >= total) return;
  int img = p / (H_LR * W_LR);
  int rem = p - img * (H_LR * W_LR);
  int i = rem >> 8;
  int j = rem & 255;

  const int ns = sp.ns;
  float wmax = -1e30f;
  for (int s = 0; s < ns; ++s) wmax = fmaxf(wmax, wts[s]);
  float e[MAX_NS];
  float denom = 0.0f;
  for (int s = 0; s < ns; ++s) { e[s] = expf(wts[s] - wmax); denom += e[s]; }
  float inv_denom = 1.0f / denom;

  const size_t imgoff = (size_t)img * (H_LR * W_LR);
  const size_t qstride = (size_t)bc * (H_LR * W_LR);
  const float* IT  = II + 0 * qstride + imgoff;
  const float* IG  = II + 1 * qstride + imgoff;
  const float* ITG = II + 2 * qstride + imgoff;
  const float* IGG = II + 3 * qstride + imgoff;

  float Aacc = 0.0f, Bacc = 0.0f;
  for (int s = 0; s < ns; ++s) {
    int kh = sp.kh[s], kw = sp.kw[s];
    int loi = i - (kh - 1) / 2; if (loi < 0) loi = 0;
    int hii = i + kh / 2 + 1;   if (hii > H_LR) hii = H_LR;
    int loj = j - (kw - 1) / 2; if (loj < 0) loj = 0;
    int hij = j + kw / 2 + 1;   if (hij > W_LR) hij = W_LR;
    float inv_n = 1.0f / (float)((hii - loi) * (hij - loj));
    int r1 = hii - 1, c1 = hij - 1;
    int r0 = loi - 1, c0 = loj - 1;   // may be -1 (no subtraction)

    auto box = [&](const float* F) -> float {
      float v = F[r1 * W_LR + c1];
      if (r0 >= 0) v -= F[r0 * W_LR + c1];
      if (c0 >= 0) v -= F[r1 * W_LR + c0];
      if (r0 >= 0 && c0 >= 0) v += F[r0 * W_LR + c0];
      return v;
    };

    float tm  = box(IT)  * inv_n;
    float gm  = box(IG)  * inv_n;
    float cov = box(ITG) * inv_n - tm * gm;
    float var = box(IGG) * inv_n - gm * gm;
    float le  = fminf(fmaxf(log_eps[s], LOG_SMALL_EPS_F), 100.0f);
    float eps = expf(le);
    float As = cov / (var + eps);
    float bs = tm - As * gm;
    float w  = e[s] * inv_denom;
    Aacc += As * w;
    Bacc += bs * w;
  }
  Ac[p] = Aacc;
  Bc[p] = Bacc;
}

// ---------------------------------------------------------------------------
// K4: fused bilinear 4x upsample of A,b + out = A*guide_hr + b
// The only HBM-heavy pass (~100 MB); A/b (6 MB) stay L2-resident.
// ---------------------------------------------------------------------------
__global__ void k_upsample(const float* __restrict__ Ac, const float* __restrict__ Bc,
                           const float* __restrict__ ghr, float* __restrict__ out,
                           int bc) {
  int p = blockIdx.x * blockDim.x + threadIdx.x;
  int img = p / (H_HR * W_HR);
  if (img >= bc) return;
  int rem = p - img * (H_HR * W_HR);
  int y = rem >> 10;   // W_HR == 1024
  int x = rem & 1023;

  float fy = (y + 0.5f) * ((float)H_LR / (float)H_HR) - 0.5f;
  float fx = (x + 0.5f) * ((float)W_LR / (float)W_HR) - 0.5f;
  int y0 = (int)floorf(fy), x0 = (int)floorf(fx);
  float ty = fy - (float)y0, tx = fx - (float)x0;
  int y0c = min(max(y0, 0), H_LR - 1), y1c = min(max(y0 + 1, 0), H_LR - 1);
  int x0c = min(max(x0, 0), W_LR - 1), x1c = min(max(x0 + 1, 0), W_LR - 1);

  size_t ib = (size_t)img * (H_LR * W_LR);
  const float* A = Ac + ib;
  const float* Bb = Bc + ib;
  float w00 = (1.0f - ty) * (1.0f - tx), w01 = (1.0f - ty) * tx;
  float w10 = ty * (1.0f - tx),          w11 = ty * tx;
  float Av = A[y0c * W_LR + x0c] * w00 + A[y0c * W_LR + x1c] * w01 +
             A[y1c * W_LR + x0c] * w10 + A[y1c * W_LR + x1c] * w11;
  float Bv = Bb[y0c * W_LR + x0c] * w00 + Bb[y0c * W_LR + x1c] * w01 +
             Bb[y1c * W_LR + x0c] * w10 + Bb[y1c * W_LR + x1c] * w11;
  out[p] = Av * ghr[p] + Bv;
}

// ---------------------------------------------------------------------------
extern "C" void kernel_launch(void* const* d_in, const int* in_sizes, int n_in,
                              void* d_out, int out_size, void* d_ws, size_t ws_size,
                              hipStream_t stream) {
  const float* guide_hr = (const float*)d_in[0];
  const float* guide    = (const float*)d_in[1];
  const float* target   = (const float*)d_in[2];
  const float* log_eps  = (const float*)d_in[3];
  const float* weights  = (const float*)d_in[4];
  float* out = (float*)d_out;

  int bc = in_sizes[1] / (H_LR * W_LR);   // B*C = 12
  int ns = in_sizes[4];
  if (ns < 1) ns = 1;
  if (ns > MAX_NS) ns = MAX_NS;

  ScaleParams sp{};
  sp.ns = ns;
  sp.kh[0] = (H_LR < 3) ? H_LR : 3;
  sp.kw[0] = (W_LR < 3) ? W_LR : 3;
  if (ns > 1) {
    int dh = (H_LR - 3) / (ns - 1);
    int dw = (W_LR - 3) / (ns - 1);
    for (int s = 1; s < ns - 1; ++s) {
      int kh = 3 + s * dh; sp.kh[s] = (kh < H_LR) ? kh : H_LR;
      int kw = 3 + s * dw; sp.kw[s] = (kw < W_LR) ? kw : W_LR;
    }
    sp.kh[ns - 1] = H_LR;
    sp.kw[ns - 1] = W_LR;
  }

  float* wsf = (float*)d_ws;
  const size_t lr = (size_t)H_LR * W_LR;
  float* II = wsf;                       // 4 * bc * lr floats (integral images)
  float* Ac = wsf + 4 * (size_t)bc * lr; // bc * lr floats
  float* Bc = Ac + (size_t)bc * lr;      // bc * lr floats

  const int nblk_scan = 4 * bc * (H_LR / 16);     // = 768 waves
  k_cumsum_w<<<nblk_scan, dim3(32), 0, stream>>>(target, guide, II, bc);
  k_cumsum_h<<<4 * bc * (W_LR / 16), dim3(32), 0, stream>>>(II, bc);

  int totalLR = bc * (int)lr;
  k_stats<<<(totalLR + 255) / 256, 256, 0, stream>>>(II, log_eps, weights, Ac, Bc, bc, sp);

  int totalHR = bc * H_HR * W_HR;
  k_upsample<<<(totalHR + 255) / 256, 256, 0, stream>>>(Ac, Bc, guide_hr, out, bc);
}